// Embedding_39737037423043
// MI455X (gfx1250) — compile-verified
//
#include <hip/hip_runtime.h>
#include <cstddef>

typedef __attribute__((ext_vector_type(2))) float v2f;
typedef __attribute__((ext_vector_type(8))) float v8f;
typedef __attribute__((ext_vector_type(4))) unsigned int v4u;
typedef __attribute__((ext_vector_type(8))) int v8i;
typedef __attribute__((ext_vector_type(4))) int v4i;

#define BN_EPS 1e-5f

// D(16x16,f32) = A(16x4,f32) x B(4x16,f32) + C  -- exact f32 matrix op on CDNA5.
// Fragment layouts (wave32): A: M=lane&15, K = v + 2*(lane>>4)
//                            B: N=lane&15, K = v + 2*(lane>>4)
//                            D: N=lane&15, M = r + 8*(lane>>4) for VGPR r
static __device__ __forceinline__ v8f wmma4(v2f a, v2f b, v8f c) {
  return __builtin_amdgcn_wmma_f32_16x16x4_f32(
      /*neg_a=*/false, a, /*neg_b=*/false, b,
      /*c_mod=*/(short)0, c, /*reuse_a=*/false, /*reuse_b=*/false);
}

// TDM gather: pull 16 indexed rows (64 f32 each) of xnT[b] into LDS, with
// 4-dword padding after each 64-dword row => staged row stride = 68 floats
// (bank-conflict-free B-fragment reads). Tracked by TENSORcnt.
static __device__ __forceinline__ void tdm_gather16(unsigned lds_off,
                                                    unsigned long long gaddr,
                                                    int N, int gidx) {
  v4u g0;
  g0[0] = 0x80000001u;                       // count=1 | gather_mode | 16b idx
  g0[1] = lds_off;                           // LDS byte offset of staging buf
  g0[2] = (unsigned)(gaddr & 0xFFFFFFFFull); // global_addr[31:0]
  g0[3] = (unsigned)((gaddr >> 32) & 0x01FFFFFFull) | (2u << 30); // [56:32]|type=2
  v8i g1;
  g1[0] = 0x07520000;              // data_size=4B | pad_en | ivl=64dw | amt=4dw
  g1[1] = (64 << 16);              // tensor_dim0 = 64 (bits 79:48)
  g1[2] = (N & 0xFFFF) << 16;      // tensor_dim1[15:0]   (bits 111:80)
  g1[3] = ((N >> 16) & 0xFFFF) | (64 << 16);  // tensor_dim1[31:16] | tile_dim0=64
  g1[4] = 16;                      // tile_dim1 = 16 valid indices
  g1[5] = 64;                      // tensor_dim0_stride = 64
  g1[6] = 0;
  g1[7] = 0;
  unsigned p[8];
#pragma unroll
  for (int w = 0; w < 8; ++w) {
    unsigned lo = (unsigned)__builtin_amdgcn_readlane(gidx, 2 * w) & 0xFFFFu;
    unsigned hi = (unsigned)__builtin_amdgcn_readlane(gidx, 2 * w + 1) & 0xFFFFu;
    p[w] = lo | (hi << 16);
  }
  v4i g2, g3;
  g2[0] = (int)p[0]; g2[1] = (int)p[1]; g2[2] = (int)p[2]; g2[3] = (int)p[3];
  g3[0] = (int)p[4]; g3[1] = (int)p[5]; g3[2] = (int)p[6]; g3[3] = (int)p[7];
  v8i g4;  // reserved/extended descriptor words (6-arg toolchain form): zeros
  g4[0] = 0; g4[1] = 0; g4[2] = 0; g4[3] = 0;
  g4[4] = 0; g4[5] = 0; g4[6] = 0; g4[7] = 0;
  __builtin_amdgcn_tensor_load_to_lds(g0, g1, g2, g3, g4, 0);
}

// ---------------------------------------------------------------------------
// Kernel 1: xnT[b][n][c] = bn0(x[b][c][n])
// ---------------------------------------------------------------------------
__global__ void bn_transpose(const float* __restrict__ x,
                             const float* __restrict__ g,
                             const float* __restrict__ bb,
                             const float* __restrict__ m,
                             const float* __restrict__ v,
                             float* __restrict__ xnT, int N) {
  __shared__ float tile[64][65];
  const int b  = blockIdx.y;
  const int n0 = blockIdx.x * 64;
  const int tx = threadIdx.x, ty = threadIdx.y;  // 16x16

  for (int c = ty; c < 64; c += 16) {
    float s = g[c] * rsqrtf(v[c] + BN_EPS);
    float t = bb[c] - m[c] * s;
    for (int nn = tx; nn < 64; nn += 16) {
      int n = n0 + nn;
      int nc = (n < N) ? n : (N - 1);
      tile[nn][c] = x[((size_t)b * 64 + c) * N + nc] * s + t;
    }
  }
  __syncthreads();
  for (int nn = ty; nn < 64; nn += 16) {
    int n = n0 + nn;
    if (n >= N) continue;
    for (int c = tx; c < 64; c += 16)
      xnT[((size_t)b * N + n) * 64 + c] = tile[nn][c];
  }
}

// ---------------------------------------------------------------------------
// Kernel 2: fold BN/bias terms into weights (see round-0 derivation).
// ---------------------------------------------------------------------------
__global__ void prep_weights(const float* __restrict__ w1,
                             const float* __restrict__ w2a,
                             const float* __restrict__ w2b,
                             const float* __restrict__ wf,
                             const float* __restrict__ b1,
                             const float* __restrict__ bfv,
                             const float* __restrict__ g2a, const float* __restrict__ b2a,
                             const float* __restrict__ m2a, const float* __restrict__ v2a,
                             const float* __restrict__ g2b, const float* __restrict__ b2b,
                             const float* __restrict__ m2b, const float* __restrict__ v2b,
                             float* __restrict__ w2aF, float* __restrict__ w2bF,
                             float* __restrict__ wcomb, float* __restrict__ cAll,
                             float* __restrict__ bffF) {
  const int o = blockIdx.x;    // 64 blocks
  const int c = threadIdx.x;   // 64 threads
  __shared__ float red[64];

  float s2a_c = g2a[c] * rsqrtf(v2a[c] + BN_EPS);
  float s2b_c = g2b[c] * rsqrtf(v2b[c] + BN_EPS);
  w2aF[o * 64 + c] = w2a[o * 64 + c] * s2a_c;
  w2bF[o * 64 + c] = w2b[o * 64 + c] * s2b_c;

  float acc = 0.f;
  for (int mm = 0; mm < 64; ++mm) acc += wf[o * 128 + mm] * w1[mm * 64 + c];
  wcomb[o * 64 + c] = acc;

  float t2a_c = b2a[c] - m2a[c] * s2a_c;
  red[c] = w2a[o * 64 + c] * t2a_c;
  __syncthreads();
  if (c == 0) {
    float s = 0.f;
    for (int i = 0; i < 64; ++i) s += red[i];
    float s2b_o = g2b[o] * rsqrtf(v2b[o] + BN_EPS);
    float t2b_o = b2b[o] - m2b[o] * s2b_o;
    cAll[o] = s + t2b_o / s2b_o;
  }
  __syncthreads();
  red[c] = wf[o * 128 + c] * b1[c];
  __syncthreads();
  if (c == 0) {
    float s = 0.f;
    for (int i = 0; i < 64; ++i) s += red[i];
    bffF[o] = bfv[o] + s;
  }
}

// ---------------------------------------------------------------------------
// Kernel 3: main EdgeConv. One wave == one (batch, 16-point) tile.
// TDM double-buffers the neighbor gather while WMMA pipelines run.
// ---------------------------------------------------------------------------
__global__ __launch_bounds__(128) void edge_main(
    const float* __restrict__ xnT, const int* __restrict__ nbr,
    const float* __restrict__ w2aF, const float* __restrict__ w2bF,
    const float* __restrict__ wcomb, const float* __restrict__ wf,
    const float* __restrict__ cAll, const float* __restrict__ bffF,
    float* __restrict__ out, int N, int tiles_per_b, int total_tiles) {
  __shared__ float sw2a[64 * 68];    // stride 68: conflict-free A-frag reads
  __shared__ float sw2b[64 * 68];
  __shared__ float rt[4 * 64 * 19];  // per-wave D->B relayout tile, stride 19
  __shared__ float stg[4][2][17 * 64];  // per-wave double-buffered TDM staging
                                        // (16 rows * 68 floats = 1088)

  for (int i = threadIdx.x; i < 4096; i += 128) {
    int o = i >> 6, c = i & 63;
    sw2a[o * 68 + c] = w2aF[i];
    sw2b[o * 68 + c] = w2bF[i];
  }
  __syncthreads();

  const int wave = threadIdx.x >> 5;
  const int lane = threadIdx.x & 31;
  const int gtile = blockIdx.x * 4 + wave;
  if (gtile >= total_tiles) return;  // wave-uniform: EXEC stays all-1s below

  const int b  = gtile / tiles_per_b;
  const int n0 = (gtile - b * tiles_per_b) * 16;
  const int h  = lane >> 4;   // lane half
  const int j  = lane & 15;   // matrix col (B/D) and row (A)
  int col = n0 + j;
  if (col >= N) col = N - 1;  // clamp; clamped lanes not stored

  float* st0 = &stg[wave][0][0];
  float* st1 = &stg[wave][1][0];
  const unsigned stoff0 = (unsigned)(size_t)(void*)st0;  // low 32b of flat = LDS offset
  const unsigned stoff1 = (unsigned)(size_t)(void*)st1;
  const unsigned long long gaddr =
      (unsigned long long)(size_t)(xnT + (size_t)b * N * 64);

  // center columns as B-fragments (reused by diff and by wcomb GEMM)
  const float* cbase = xnT + ((size_t)b * N + col) * 64 + 2 * h;
  v2f cx[16];
#pragma unroll
  for (int kc = 0; kc < 16; ++kc) cx[kc] = *(const v2f*)(cbase + kc * 4);

  // per-lane bias constants for relu stage (o = 16*ot + 8*h + r)
  float callv[32];
#pragma unroll
  for (int ot = 0; ot < 4; ++ot)
#pragma unroll
    for (int r = 0; r < 8; ++r) callv[ot * 8 + r] = cAll[16 * ot + 8 * h + r];

  v8f mx[4];
#pragma unroll
  for (int ot = 0; ot < 4; ++ot)
#pragma unroll
    for (int r = 0; r < 8; ++r) mx[ot][r] = -3.0e38f;

  float* rw = rt + wave * (64 * 19);

  // prime the pipeline: TDM gather for k=1 into buffer 1
  {
    int g1i = nbr[((size_t)b * 16 + 1) * N + col];
    tdm_gather16(stoff1, gaddr, N, g1i);
  }

  for (int k = 1; k < 16; ++k) {
    // issue DMA for k+1, then wait only for k's buffer
    if (k < 15) {
      int gn = nbr[((size_t)b * 16 + (k + 1)) * N + col];
      tdm_gather16((k & 1) ? stoff0 : stoff1, gaddr, N, gn);
      __builtin_amdgcn_s_wait_tensorcnt(1);
    } else {
      __builtin_amdgcn_s_wait_tensorcnt(0);
    }
    const float* stage = (k & 1) ? st1 : st0;
    const float* srow = stage + j * 68 + 2 * h;  // this lane's staged column

    // GEMM 1: h2 = w2aF @ (gathered - center)
    v8f acc[4] = {};
#pragma unroll
    for (int kc = 0; kc < 16; ++kc) {
      v2f gg = *(const v2f*)(srow + kc * 4);
      v2f df = gg - cx[kc];
      const float* arow = sw2a + kc * 4 + 2 * h;
#pragma unroll
      for (int ot = 0; ot < 4; ++ot) {
        v2f a = *(const v2f*)(arow + (16 * ot + j) * 68);
        acc[ot] = wmma4(a, df, acc[ot]);
      }
    }

    // bias + relu, relayout D->B via per-wave LDS tile
#pragma unroll
    for (int ot = 0; ot < 4; ++ot)
#pragma unroll
      for (int r = 0; r < 8; ++r) {
        float u = fmaxf(acc[ot][r] + callv[ot * 8 + r], 0.f);
        rw[(16 * ot + 8 * h + r) * 19 + j] = u;
      }
    asm volatile("s_wait_dscnt 0" ::: "memory");

    // GEMM 2: h3 = w2bF @ relu(...)
    v8f acc2[4] = {};
#pragma unroll
    for (int kc = 0; kc < 16; ++kc) {
      v2f bf2;
      bf2.x = rw[(kc * 4 + 2 * h) * 19 + j];
      bf2.y = rw[(kc * 4 + 2 * h + 1) * 19 + j];
      const float* brow = sw2b + kc * 4 + 2 * h;
#pragma unroll
      for (int ot = 0; ot < 4; ++ot) {
        v2f a = *(const v2f*)(brow + (16 * ot + j) * 68);
        acc2[ot] = wmma4(a, bf2, acc2[ot]);
      }
    }
#pragma unroll
    for (int ot = 0; ot < 4; ++ot)
#pragma unroll
      for (int r = 0; r < 8; ++r) mx[ot][r] = fmaxf(mx[ot][r], acc2[ot][r]);
    asm volatile("s_wait_dscnt 0" ::: "memory");
  }

  // neigh (max) -> B-fragments via LDS
#pragma unroll
  for (int ot = 0; ot < 4; ++ot)
#pragma unroll
    for (int r = 0; r < 8; ++r) rw[(16 * ot + 8 * h + r) * 19 + j] = mx[ot][r];
  asm volatile("s_wait_dscnt 0" ::: "memory");

  // out = wcomb @ xn_center + wf[:,64:] @ neigh + bffF
  v8f oacc[4] = {};
#pragma unroll
  for (int kc = 0; kc < 16; ++kc) {
    v2f nf;
    nf.x = rw[(kc * 4 + 2 * h) * 19 + j];
    nf.y = rw[(kc * 4 + 2 * h + 1) * 19 + j];
#pragma unroll
    for (int ot = 0; ot < 4; ++ot) {
      v2f awc = *(const v2f*)(wcomb + (size_t)(16 * ot + j) * 64 + kc * 4 + 2 * h);
      oacc[ot] = wmma4(awc, cx[kc], oacc[ot]);
      v2f awn = *(const v2f*)(wf + (size_t)(16 * ot + j) * 128 + 64 + kc * 4 + 2 * h);
      oacc[ot] = wmma4(awn, nf, oacc[ot]);
    }
  }

  if (n0 + j < N) {
#pragma unroll
    for (int ot = 0; ot < 4; ++ot)
#pragma unroll
      for (int r = 0; r < 8; ++r) {
        int o = 16 * ot + 8 * h + r;
        out[((size_t)b * 64 + o) * N + n0 + j] = oacc[ot][r] + bffF[o];
      }
  }
}

// ---------------------------------------------------------------------------
extern "C" void kernel_launch(void* const* d_in, const int* in_sizes, int n_in,
                              void* d_out, int out_size, void* d_ws, size_t ws_size,
                              hipStream_t stream) {
  const float* x    = (const float*)d_in[0];
  const int*   nbr  = (const int*)d_in[1];
  const float* bn0g = (const float*)d_in[2];
  const float* bn0b = (const float*)d_in[3];
  const float* bn0m = (const float*)d_in[4];
  const float* bn0v = (const float*)d_in[5];
  const float* w1   = (const float*)d_in[6];
  const float* b1   = (const float*)d_in[7];
  const float* g2a  = (const float*)d_in[8];
  const float* b2a  = (const float*)d_in[9];
  const float* m2a  = (const float*)d_in[10];
  const float* v2a  = (const float*)d_in[11];
  const float* w2a  = (const float*)d_in[12];
  const float* g2b  = (const float*)d_in[13];
  const float* b2b  = (const float*)d_in[14];
  const float* m2b  = (const float*)d_in[15];
  const float* v2b  = (const float*)d_in[16];
  const float* w2b  = (const float*)d_in[17];
  const float* wf   = (const float*)d_in[18];
  const float* bfv  = (const float*)d_in[19];

  const int B = 2, C = 64;
  const int N = in_sizes[0] / (B * C);

  float* ws = (float*)d_ws;
  size_t off = 0;
  float* xnT   = ws + off; off += (size_t)B * N * C;
  float* w2aF  = ws + off; off += 4096;
  float* w2bF  = ws + off; off += 4096;
  float* wcomb = ws + off; off += 4096;
  float* cAll  = ws + off; off += 64;
  float* bffF  = ws + off; off += 64;

  dim3 tb(16, 16);
  dim3 tg((N + 63) / 64, B);
  bn_transpose<<<tg, tb, 0, stream>>>(x, bn0g, bn0b, bn0m, bn0v, xnT, N);

  prep_weights<<<64, 64, 0, stream>>>(w1, w2a, w2b, wf, b1, bfv,
                                      g2a, b2a, m2a, v2a,
                                      g2b, b2b, m2b, v2b,
                                      w2aF, w2bF, wcomb, cAll, bffF);

  int tiles_per_b = (N + 15) / 16;
  int total = B * tiles_per_b;
  int blocks = (total + 3) / 4;
  edge_main<<<blocks, 128, 0, stream>>>(xnT, nbr, w2aF, w2bF, wcomb, wf,
                                        cAll, bffF, (float*)d_out,
                                        N, tiles_per_b, total);
}